// SetCriterion_81217831567395
// MI455X (gfx1250) — compile-verified
//
#include <hip/hip_runtime.h>
#include <math.h>

#define B_      8
#define Q_      1000
#define P_      16
#define C_      1
#define D_      256
#define K_      50
#define F_      8
#define N_SEL   (B_ * K_)   /* 400  */
#define N_ROWS  (B_ * Q_)   /* 8000 */
#define ALPHA_  0.25f
#define TEMP_INV 2.0f       /* 1 / 0.5 */
#define IOU_THR_ 0.4f
#define EPS_    1e-9f

#define NTILES_N 500  /* 8000/16 */
#define NTILES_M 25   /* 400/16  */
#define FOCAL_BLOCKS ((B_ * Q_ * P_ + 255) / 256)   /* 500 */
#define COUNT_BLOCKS ((N_ROWS + 255) / 256)         /* 32  */

typedef float v2f __attribute__((ext_vector_type(2)));
typedef float v8f __attribute__((ext_vector_type(8)));

// ---------------- workspace layout (float offsets) ----------------
#define NEG_OFF   0                            // [8000*256]
#define SEL_OFF   (NEG_OFF   + N_ROWS * D_)    // [400*256]
#define W_OFF     (SEL_OFF   + N_SEL * D_)     // [8000]
#define SIM_OFF   (W_OFF     + N_ROWS)         // [400*400]
#define NSUM_OFF  (SIM_OFF   + N_SEL * N_SEL)  // [400]
#define PART_OFF  (NSUM_OFF  + N_SEL)          // [500*400] per-tile row sums
#define FPART_OFF (PART_OFF  + NTILES_N * N_SEL) // [500] focal block partials
#define WPART_OFF (FPART_OFF + FOCAL_BLOCKS)   // [32] w_sum block partials
#define TC_OFF    (WPART_OFF + COUNT_BLOCKS)   // [8000] ints

// ---------------- init: tc = C (unmatched default) ----------------
__global__ void init_kernel(int* __restrict__ tc) {
    int idx = blockIdx.x * blockDim.x + threadIdx.x;
    if (idx < N_ROWS) tc[idx] = C_;
}

// ---------------- scatter matched labels into tc ----------------
__global__ void scatter_kernel(const int* __restrict__ midx,
                               const int* __restrict__ tlab,
                               int* __restrict__ tc) {
    int idx = blockIdx.x * blockDim.x + threadIdx.x;
    if (idx >= N_SEL) return;
    int b = idx / K_;
    int q = midx[idx];
    tc[b * Q_ + q] = tlab[idx];
}

// ------------- sigmoid focal loss -> per-block partials -------------
__global__ void focal_kernel(const float* __restrict__ logits,
                             const int* __restrict__ tc,
                             float* __restrict__ fpart) {
    __shared__ float red[256];
    int idx = blockIdx.x * blockDim.x + threadIdx.x; // over B*Q*P (C==1)
    float v = 0.f;
    if (idx < B_ * Q_ * P_) {
        int row = idx / P_;                    // b*Q + q
        float x = logits[idx];
        float t = (tc[row] == 0) ? 1.f : 0.f;  // C==1: class-0 one-hot
        float prob = 1.f / (1.f + expf(-x));
        float ce = fmaxf(x, 0.f) - x * t + log1pf(expf(-fabsf(x)));
        float p_t = prob * t + (1.f - prob) * (1.f - t);
        float omp = 1.f - p_t;
        v = (ALPHA_ * t + (1.f - ALPHA_) * (1.f - t)) * ce * omp * omp;
    }
    red[threadIdx.x] = v;
    __syncthreads();
    for (int off = 128; off > 0; off >>= 1) {
        if (threadIdx.x < off) red[threadIdx.x] += red[threadIdx.x + off];
        __syncthreads();
    }
    if (threadIdx.x == 0) fpart[blockIdx.x] = red[0];
}

// ---------------- emb_mean over P + L2-normalize -> neg ----------------
__global__ void embmean_kernel(const float* __restrict__ emb,
                               float* __restrict__ neg) {
    int row = blockIdx.x;     // 0..7999
    int d   = threadIdx.x;    // 0..255
    const float* base = emb + (size_t)row * P_ * D_ + d;
    float s = 0.f;
#pragma unroll
    for (int p = 0; p < P_; ++p) s += base[p * D_];
    s *= (1.f / P_);
    __shared__ float red[256];
    red[d] = s * s;
    __syncthreads();
    for (int off = 128; off > 0; off >>= 1) {
        if (d < off) red[d] += red[d + off];
        __syncthreads();
    }
    float norm = sqrtf(red[0]);
    neg[(size_t)row * D_ + d] = s / norm;
}

// ---------------- gather normalized matched rows -> sel ----------------
__global__ void selgather_kernel(const float* __restrict__ neg,
                                 const int* __restrict__ midx,
                                 float* __restrict__ sel) {
    int idx = blockIdx.x * blockDim.x + threadIdx.x;
    if (idx >= N_SEL * D_) return;
    int i = idx / D_, d = idx % D_;
    int b = i / K_;
    int q = midx[i];
    sel[idx] = neg[((size_t)(b * Q_ + q)) * D_ + d];
}

// ----------- GIoU counts -> w, plus per-block w_sum partials -----------
__global__ void counts_kernel(const float* __restrict__ ctrl,
                              const float* __restrict__ fp,
                              const int* __restrict__ tc,
                              float* __restrict__ w,
                              float* __restrict__ wpart) {
    __shared__ float red[256];
    int row = blockIdx.x * blockDim.x + threadIdx.x;
    float cnt = 0.f;
    if (row < N_ROWS) {
        const float* cp = ctrl + (size_t)row * P_ * 2;
        float mnx = cp[0], mny = cp[1], mxx = cp[0], mxy = cp[1];
#pragma unroll
        for (int p = 1; p < P_; ++p) {
            mnx = fminf(mnx, cp[2 * p]);     mxx = fmaxf(mxx, cp[2 * p]);
            mny = fminf(mny, cp[2 * p + 1]); mxy = fmaxf(mxy, cp[2 * p + 1]);
        }
        if (tc[row] == C_) {  // unmatched query
            int b = row / Q_;
            const float* fb = fp + (size_t)b * F_ * 4;
            float a1 = (mxx - mnx) * (mxy - mny);
#pragma unroll
            for (int f = 0; f < F_; ++f) {
                float x0 = fb[4 * f + 0], y0 = fb[4 * f + 1];
                float x1 = fb[4 * f + 2], y1 = fb[4 * f + 3];
                float ix1 = fmaxf(mnx, x0), iy1 = fmaxf(mny, y0);
                float ix2 = fminf(mxx, x1), iy2 = fminf(mxy, y1);
                float inter = fmaxf(ix2 - ix1, 0.f) * fmaxf(iy2 - iy1, 0.f);
                float a2  = (x1 - x0) * (y1 - y0);
                float uni = a1 + a2 - inter;
                float iou = inter / (uni + EPS_);
                float hx1 = fminf(mnx, x0), hy1 = fminf(mny, y0);
                float hx2 = fmaxf(mxx, x1), hy2 = fmaxf(mxy, y1);
                float hull = (hx2 - hx1) * (hy2 - hy1);
                float g = iou - (hull - uni) / (hull + EPS_);
                if (g > IOU_THR_) cnt += 1.f;
            }
        }
        w[row] = cnt;
    }
    red[threadIdx.x] = cnt;
    __syncthreads();
    for (int off = 128; off > 0; off >>= 1) {
        if (threadIdx.x < off) red[threadIdx.x] += red[threadIdx.x + off];
        __syncthreads();
    }
    if (threadIdx.x == 0) wpart[blockIdx.x] = red[0];
}

// ---- WMMA GEMM: partial[nt*400+i] = sum_{j in tile nt} exp(2*dot)*w[j] ----
__global__ void __launch_bounds__(256)
gemm_negsum_kernel(const float* __restrict__ sel, const float* __restrict__ neg,
                   const float* __restrict__ w, float* __restrict__ partial) {
    int wave = (blockIdx.x * blockDim.x + threadIdx.x) >> 5;   // wave-uniform
    int lane = threadIdx.x & 31;
    if (wave >= NTILES_M * NTILES_N) return;                    // uniform exit
    int mt = wave / NTILES_N, nt = wave % NTILES_N;
    int m = mt * 16 + (lane & 15);
    int n = nt * 16 + (lane & 15);
    int koff = (lane < 16) ? 0 : 2;
    const float* arow = sel + (size_t)m * D_ + koff;
    const float* brow = neg + (size_t)n * D_ + koff;
    v8f c = {};
#pragma unroll 8
    for (int k = 0; k < D_; k += 4) {
        v2f a = *(const v2f*)(arow + k);
        v2f b = *(const v2f*)(brow + k);
        c = __builtin_amdgcn_wmma_f32_16x16x4_f32(
                false, a, false, b, (short)0, c, false, false);
    }
    float wj = w[n];
    int m0 = mt * 16 + ((lane < 16) ? 0 : 8);
#pragma unroll
    for (int r = 0; r < 8; ++r) {
        float v = expf(TEMP_INV * c[r]) * wj;   // exp(sim/TEMP) * weight
        v += __shfl_xor(v, 1);
        v += __shfl_xor(v, 2);
        v += __shfl_xor(v, 4);
        v += __shfl_xor(v, 8);                  // sum over 16 N-columns
        if ((lane & 15) == 0) partial[(size_t)nt * N_SEL + m0 + r] = v;
    }
}

// ----- deterministic reduce: neg_sum[m] = sum_nt partial[nt*400+m] -----
__global__ void negsum_reduce_kernel(const float* __restrict__ partial,
                                     float* __restrict__ neg_sum) {
    int m = blockIdx.x * blockDim.x + threadIdx.x;
    if (m >= N_SEL) return;
    float s = 0.f;
    for (int nt = 0; nt < NTILES_N; ++nt)       // fixed order, coalesced
        s += partial[(size_t)nt * N_SEL + m];
    neg_sum[m] = s;
}

// ---------------- WMMA GEMM: sim = 2 * sel @ sel^T ----------------
__global__ void __launch_bounds__(256)
gemm_sim_kernel(const float* __restrict__ sel, float* __restrict__ sim) {
    int wave = (blockIdx.x * blockDim.x + threadIdx.x) >> 5;
    int lane = threadIdx.x & 31;
    if (wave >= NTILES_M * NTILES_M) return;
    int mt = wave / NTILES_M, nt = wave % NTILES_M;
    int m = mt * 16 + (lane & 15);
    int n = nt * 16 + (lane & 15);
    int koff = (lane < 16) ? 0 : 2;
    const float* arow = sel + (size_t)m * D_ + koff;
    const float* brow = sel + (size_t)n * D_ + koff;
    v8f c = {};
#pragma unroll 8
    for (int k = 0; k < D_; k += 4) {
        v2f a = *(const v2f*)(arow + k);
        v2f b = *(const v2f*)(brow + k);
        c = __builtin_amdgcn_wmma_f32_16x16x4_f32(
                false, a, false, b, (short)0, c, false, false);
    }
    int m0 = mt * 16 + ((lane < 16) ? 0 : 8);
    int nc = nt * 16 + (lane & 15);
#pragma unroll
    for (int r = 0; r < 8; ++r)
        sim[(size_t)(m0 + r) * N_SEL + nc] = TEMP_INV * c[r];
}

// ---------------- final: pair loss + deterministic totals ----------------
__global__ void final_kernel(const float* __restrict__ sim,
                             const float* __restrict__ neg_sum,
                             const int* __restrict__ labels,
                             const float* __restrict__ fpart,
                             const float* __restrict__ wpart,
                             const int* __restrict__ num_inst_p,
                             float* __restrict__ out) {
    __shared__ float reds[256];
    __shared__ float redn[256];
    __shared__ float redf[256];
    __shared__ float redw[256];
    int tid = threadIdx.x;
    float s = 0.f, np = 0.f;
    for (int p = tid; p < N_SEL * N_SEL; p += blockDim.x) {
        int i = p / N_SEL, k = p % N_SEL;
        if (i != k && labels[i] == labels[k]) {
            float sv = sim[p];
            float es = expf(sv);
            s += -logf(es / (es + neg_sum[i] + 1e-6f));
            np += 1.f;
        }
    }
    float f = 0.f;
    for (int i = tid; i < FOCAL_BLOCKS; i += blockDim.x) f += fpart[i];
    float wv = (tid < COUNT_BLOCKS) ? wpart[tid] : 0.f;
    reds[tid] = s; redn[tid] = np; redf[tid] = f; redw[tid] = wv;
    __syncthreads();
    for (int off = 128; off > 0; off >>= 1) {
        if (tid < off) {
            reds[tid] += reds[tid + off];
            redn[tid] += redn[tid + off];
            redf[tid] += redf[tid + off];
            redw[tid] += redw[tid + off];
        }
        __syncthreads();
    }
    if (tid == 0) {
        float ninst = (float)num_inst_p[0];
        out[0] = redf[0] / ((float)P_ * ninst);  // Σloss/(Q*P)/ninst*Q
        float npos = redn[0], wsum = redw[0];
        out[1] = (wsum > 0.f && npos > 0.f) ? reds[0] / fmaxf(npos, 1.f) : 0.f;
    }
}

extern "C" void kernel_launch(void* const* d_in, const int* in_sizes, int n_in,
                              void* d_out, int out_size, void* d_ws, size_t ws_size,
                              hipStream_t stream) {
    const float* pred_logits = (const float*)d_in[0];
    const float* query_embed = (const float*)d_in[1];
    const float* pred_ctrl   = (const float*)d_in[2];
    const float* fp_bboxes   = (const float*)d_in[3];
    const int*   matched_idx = (const int*)d_in[4];
    const int*   target_lab  = (const int*)d_in[5];
    const int*   num_inst    = (const int*)d_in[6];
    float* out = (float*)d_out;

    float* ws      = (float*)d_ws;
    float* neg     = ws + NEG_OFF;
    float* sel     = ws + SEL_OFF;
    float* w       = ws + W_OFF;
    float* sim     = ws + SIM_OFF;
    float* neg_sum = ws + NSUM_OFF;
    float* partial = ws + PART_OFF;
    float* fpart   = ws + FPART_OFF;
    float* wpart   = ws + WPART_OFF;
    int*   tc      = (int*)(ws + TC_OFF);

    init_kernel<<<(N_ROWS + 255) / 256, 256, 0, stream>>>(tc);
    scatter_kernel<<<(N_SEL + 255) / 256, 256, 0, stream>>>(matched_idx, target_lab, tc);
    focal_kernel<<<FOCAL_BLOCKS, 256, 0, stream>>>(pred_logits, tc, fpart);
    embmean_kernel<<<N_ROWS, 256, 0, stream>>>(query_embed, neg);
    selgather_kernel<<<(N_SEL * D_ + 255) / 256, 256, 0, stream>>>(neg, matched_idx, sel);
    counts_kernel<<<COUNT_BLOCKS, 256, 0, stream>>>(pred_ctrl, fp_bboxes, tc, w, wpart);

    int waves1 = NTILES_M * NTILES_N;                    // 12500 tiles
    gemm_negsum_kernel<<<(waves1 * 32 + 255) / 256, 256, 0, stream>>>(sel, neg, w, partial);
    negsum_reduce_kernel<<<(N_SEL + 255) / 256, 256, 0, stream>>>(partial, neg_sum);
    int waves2 = NTILES_M * NTILES_M;                    // 625 tiles
    gemm_sim_kernel<<<(waves2 * 32 + 255) / 256, 256, 0, stream>>>(sel, sim);

    final_kernel<<<1, 256, 0, stream>>>(sim, neg_sum, target_lab, fpart, wpart, num_inst, out);
}